// EvalWrapper_6004364280777
// MI455X (gfx1250) — compile-verified
//
#include <hip/hip_runtime.h>
#include <hip/hip_bf16.h>
#include <math.h>

// Problem constants (from the reference)
#define B_   2
#define H_   96
#define W_   128          // power of two: x = p & 127, y = p >> 7
#define VN_  9
#define NH_  128
#define P_   (H_*W_)      // 12288
#define CH_  2048         // pixels staged in LDS per chunk (P_/CH_ = 6 chunks)
#define THRESH_ 0.99f
#define EPS_    1e-6f
#define BIG_    3.0e38f   // "mask off" threshold: no finite dot can pass

typedef __attribute__((ext_vector_type(2))) float v2f;
typedef __attribute__((ext_vector_type(8))) float v8f;

#if defined(__gfx1250__) && __has_builtin(__builtin_amdgcn_wmma_f32_16x16x4_f32)
#define USE_WMMA_F32 1
#else
#define USE_WMMA_F32 0
#endif

// Async global->LDS (CDNA5, ASYNCcnt-tracked) — probe-guarded.
// Signature (from round-2 diagnostic + LLVM def "vi*1i*3IiIi"):
//   void __builtin_amdgcn_global_load_async_to_lds_b32(
//       int addrspace(1)* gsrc, int addrspace(3)* ldst, Imm int offset, Imm int cpol)
#if defined(__gfx1250__) && \
    __has_builtin(__builtin_amdgcn_global_load_async_to_lds_b32) && \
    __has_builtin(__builtin_amdgcn_s_wait_asynccnt)
#define USE_ASYNC_LDS 1
typedef __attribute__((address_space(1))) int* as1_i32;
typedef __attribute__((address_space(3))) int* as3_i32;
#else
#define USE_ASYNC_LDS 0
#endif

// Raw hardware sqrt (single v_sqrt_f32 TRANS op, ~1 ulp; co-executes with WMMA)
#if defined(__AMDGCN__) && __has_builtin(__builtin_amdgcn_sqrtf)
#define FAST_SQRTF(x) __builtin_amdgcn_sqrtf(x)
#else
#define FAST_SQRTF(x) sqrtf(x)
#endif

// ---------------------------------------------------------------------------
// Kernel 1: hypothesis generation — one thread per (b, v, h) line intersection
// ---------------------------------------------------------------------------
__global__ __launch_bounds__(256) void hyp_gen_kernel(
    const int* __restrict__ sample_idx,   // [B, VN, NH, 2]
    const float* __restrict__ vp,         // [B, 2*VN, H, W]
    float* __restrict__ hyp_x, float* __restrict__ hyp_y,
    int* __restrict__ hvalid) {
  int tid = blockIdx.x * blockDim.x + threadIdx.x;
  if (tid >= B_ * VN_ * NH_) return;
  int b = tid / (VN_ * NH_);
  int r = tid % (VN_ * NH_);
  int v = r / NH_;

  int i0 = sample_idx[tid * 2 + 0];
  int i1 = sample_idx[tid * 2 + 1];
  float x0 = (float)(i0 & (W_ - 1)), y0 = (float)(i0 >> 7);
  float x1 = (float)(i1 & (W_ - 1)), y1 = (float)(i1 >> 7);

  const float* ch = vp + (size_t)b * 2 * VN_ * P_ + (size_t)(2 * v) * P_;
  float d0x = ch[i0], d0y = ch[P_ + i0];
  float d1x = ch[i1], d1y = ch[P_ + i1];

  float den = d0x * d1y - d0y * d1x;
  int val = fabsf(den) > EPS_;
  float t = ((x1 - x0) * d1y - (y1 - y0) * d1x) / (val ? den : 1.0f);
  hyp_x[tid] = x0 + t * d0x;
  hyp_y[tid] = y0 + t * d0y;
  hvalid[tid] = val;
}

// ---------------------------------------------------------------------------
// Kernel 2: voting (WMMA f32 16x16x4) + argmax + fused LSQ refinement.
// One block per (b, v); 8 waves, each wave owns a 16-hypothesis tile.
//
// dot[h,p]  = (hx, hy, 1, 0)          . (dirx, diry, -(px*dirx+py*diry), 0)
// len2[h,p] = (1, hx, hy, hx^2+hy^2)  . (px^2+py^2, -2px, -2py, 1)
// inlier&mask <=> dot > stdn*(sqrt(len2)+EPS), stdn = mask? 0.99*(|dir|+EPS):BIG
//
// Pixels are staged per 2048-chunk in LDS (packed float4 + threshold), so the
// per-pixel sqrt/derivations happen once per block, not once per wave. The
// staging uses GLOBAL_LOAD_ASYNC_TO_LDS_B32 (ASYNCcnt) when available.
//
// ISA layouts (§7.12.2, wave32): A 16x4: lane%16 = M, K = vgpr + 2*(lane>=16)
//   (B assumed symmetric: lane%16 = N, K = vgpr + 2*(lane>=16))
// C/D 16x16: lane holds column N = lane%16, row M = vgpr + 8*(lane>=16)
// ---------------------------------------------------------------------------
__global__ __launch_bounds__(256) void vote_refine_kernel(
    const int* __restrict__ mask,         // [B, H, W]
    const float* __restrict__ vp,         // [B, 2*VN, H, W]
    const float* __restrict__ hyp_x, const float* __restrict__ hyp_y,
    const int* __restrict__ hvalid,
    float* __restrict__ out) {            // [B, VN, 2]
  int bv = blockIdx.x;
  int b = bv / VN_, v = bv % VN_;
  int tid = threadIdx.x;
  int wave = tid >> 5, lane = tid & 31;
  int nl = lane & 15, half = lane >> 4;

  __shared__ float4 sq[CH_];      // {dirx, diry, c0, pp} per staged pixel
  __shared__ float  stdn_s[CH_];  // mask-folded threshold per staged pixel
  __shared__ int    smk[CH_];     // raw mask staging (async path)
  __shared__ float  s_counts[NH_];
  __shared__ float  s_win[2];
  __shared__ float  s_red[5][256];

  const float* dirx = vp + (size_t)b * 2 * VN_ * P_ + (size_t)(2 * v) * P_;
  const float* diry = dirx + P_;
  const int* mk = mask + (size_t)b * P_;

  int hbase = bv * NH_ + wave * 16;
  float hx = hyp_x[hbase + nl];   // lanes 0..15 hold hyp M = nl (halves mirror)
  float hy = hyp_y[hbase + nl];

  float cnt[8];
#pragma unroll
  for (int r2 = 0; r2 < 8; ++r2) cnt[r2] = 0.0f;

#if USE_WMMA_F32
  // Loop-invariant A matrices (per wave's 16-hypothesis tile)
  v2f a_dot, a_len;
  a_dot.x = half ? 1.0f : hx;                 // K0 / K2
  a_dot.y = half ? 0.0f : hy;                 // K1 / K3
  a_len.x = half ? hy : 1.0f;                 // K0 / K2
  a_len.y = half ? (hx * hx + hy * hy) : hx;  // K1 / K3
#endif

  for (int cbase = 0; cbase < P_; cbase += CH_) {
    __syncthreads();  // all waves done with previous chunk's LDS

    // ---- Phase A: stage + preprocess 2048 pixels (once per block) ----
#if USE_ASYNC_LDS
    for (int i = tid; i < CH_; i += 256) {
      int p = cbase + i;
      __builtin_amdgcn_global_load_async_to_lds_b32(
          (as1_i32)(dirx + p), (as3_i32)(&sq[i].x), 0, 0);
      __builtin_amdgcn_global_load_async_to_lds_b32(
          (as1_i32)(diry + p), (as3_i32)(&sq[i].y), 0, 0);
      __builtin_amdgcn_global_load_async_to_lds_b32(
          (as1_i32)(mk + p), (as3_i32)(&smk[i]), 0, 0);
    }
    __builtin_amdgcn_s_wait_asynccnt(0);
    __syncthreads();
    for (int i = tid; i < CH_; i += 256) {
      int p = cbase + i;
      float ddx = sq[i].x, ddy = sq[i].y;
      float px = (float)(p & (W_ - 1)), py = (float)(p >> 7);
      float dn = sqrtf(ddx * ddx + ddy * ddy) + EPS_;  // once per pixel
      stdn_s[i] = (smk[i] > 0) ? THRESH_ * dn : BIG_;
      sq[i].z = -(px * ddx + py * ddy);
      sq[i].w = px * px + py * py;
    }
#else
    for (int i = tid; i < CH_; i += 256) {
      int p = cbase + i;
      float ddx = dirx[p], ddy = diry[p];
      float px = (float)(p & (W_ - 1)), py = (float)(p >> 7);
      float dn = sqrtf(ddx * ddx + ddy * ddy) + EPS_;
      stdn_s[i] = (mk[p] > 0) ? THRESH_ * dn : BIG_;
      sq[i] = make_float4(ddx, ddy, -(px * ddx + py * ddy), px * px + py * py);
    }
#endif
    __syncthreads();

    // ---- Phase B: each wave votes its 16 hypotheses over the chunk ----
    for (int t = 0; t < CH_; t += 16) {
      int i = t + nl;                         // both halves use pixel N = nl
      float4 q = sq[i];                       // ds_load_b128
      float tdn = stdn_s[i];
      int p = cbase + i;
      float px = (float)(p & (W_ - 1));
      float py = (float)(p >> 7);
      float tdne = tdn * EPS_;

#if USE_WMMA_F32
      v2f b_dot, b_len;
      b_dot.x = half ? q.z : q.x;             // K0 / K2
      b_dot.y = half ? 0.0f : q.y;            // K1 / K3
      b_len.x = half ? (-2.0f * py) : q.w;    // K0 / K2
      b_len.y = half ? 1.0f : (-2.0f * px);   // K1 / K3
      v8f zero = {0.f, 0.f, 0.f, 0.f, 0.f, 0.f, 0.f, 0.f};
      v8f d_dot = __builtin_amdgcn_wmma_f32_16x16x4_f32(
          false, a_dot, false, b_dot, (short)0, zero, false, false);
      v8f d_len = __builtin_amdgcn_wmma_f32_16x16x4_f32(
          false, a_len, false, b_len, (short)0, zero, false, false);
#pragma unroll
      for (int r2 = 0; r2 < 8; ++r2) {
        // cancellation => l2<0 => sqrt NaN => compare false == not inlier (ok:
        // such a pixel sits at the hypothesis point, where ref cos ~ 0)
        float s = FAST_SQRTF(d_len[r2]);      // 1 TRANS op, co-executes w/ WMMA
        float rhs = fmaf(tdn, s, tdne);       // tdn*(sqrt+EPS)
        if (d_dot[r2] > rhs) cnt[r2] += 1.0f; // mask folded into tdn
      }
#else
#pragma unroll
      for (int r2 = 0; r2 < 8; ++r2) {
        int m = r2 + 8 * half;
        float hxm = __shfl(hx, m, 32);
        float hym = __shfl(hy, m, 32);
        float ex = hxm - px, ey = hym - py;
        float dotv = ex * q.x + ey * q.y;
        float s = FAST_SQRTF(ex * ex + ey * ey);
        float rhs = fmaf(tdn, s, tdne);
        if (dotv > rhs) cnt[r2] += 1.0f;
      }
#endif
    }
  }

  // Reduce over the 16 pixel columns (masks <= 8 keep halves independent)
#pragma unroll
  for (int r2 = 0; r2 < 8; ++r2) {
#pragma unroll
    for (int mm = 1; mm < 16; mm <<= 1)
      cnt[r2] += __shfl_xor(cnt[r2], mm, 32);
  }
  if (nl == 0) {  // lanes 0 (M=0..7) and 16 (M=8..15)
#pragma unroll
    for (int r2 = 0; r2 < 8; ++r2) {
      int h = wave * 16 + r2 + 8 * half;
      s_counts[h] = hvalid[bv * NH_ + h] ? cnt[r2] : -1.0f;
    }
  }
  __syncthreads();

  // argmax with first-max tie-break (matches jnp.argmax)
  if (tid == 0) {
    float bc = s_counts[0];
    int bh = 0;
    for (int h2 = 1; h2 < NH_; ++h2)
      if (s_counts[h2] > bc) { bc = s_counts[h2]; bh = h2; }
    s_win[0] = hyp_x[bv * NH_ + bh];
    s_win[1] = hyp_y[bv * NH_ + bh];
  }
  __syncthreads();

  // Weighted least squares over inliers of the winning hypothesis
  float wx = s_win[0], wy = s_win[1];
  float a00 = 0.f, a01 = 0.f, a11 = 0.f, rb0 = 0.f, rb1 = 0.f;
  for (int p = tid; p < P_; p += 256) {
    if (mk[p] <= 0) continue;                 // wpix==0 either way
    float px = (float)(p & (W_ - 1)), py = (float)(p >> 7);
    float ddx = dirx[p], ddy = diry[p];
    float dn = sqrtf(ddx * ddx + ddy * ddy) + EPS_;
    float ex = wx - px, ey = wy - py;
    float dotv = ex * ddx + ey * ddy;
    float ln = sqrtf(ex * ex + ey * ey) + EPS_;
    if (dotv > THRESH_ * ln * dn) {           // cos > THRESH (positive sides)
      float n0 = ddy / dn, n1 = -ddx / dn;    // normal = (nrm.y, -nrm.x)
      float nb = n0 * px + n1 * py;
      a00 += n0 * n0; a01 += n0 * n1; a11 += n1 * n1;
      rb0 += nb * n0; rb1 += nb * n1;
    }
  }
  // Deterministic tree reduction (no float atomics -> replay-stable output)
  s_red[0][tid] = a00; s_red[1][tid] = a01; s_red[2][tid] = a11;
  s_red[3][tid] = rb0; s_red[4][tid] = rb1;
  __syncthreads();
  for (int s = 128; s > 0; s >>= 1) {
    if (tid < s) {
#pragma unroll
      for (int k = 0; k < 5; ++k) s_red[k][tid] += s_red[k][tid + s];
    }
    __syncthreads();
  }
  if (tid == 0) {
    float A00 = s_red[0][0] + EPS_, A01 = s_red[1][0], A11 = s_red[2][0] + EPS_;
    float b0 = s_red[3][0], b1 = s_red[4][0];
    float inv = 1.0f / (A00 * A11 - A01 * A01);
    out[bv * 2 + 0] = (A11 * b0 - A01 * b1) * inv;
    out[bv * 2 + 1] = (A00 * b1 - A01 * b0) * inv;
  }
}

// ---------------------------------------------------------------------------
extern "C" void kernel_launch(void* const* d_in, const int* in_sizes, int n_in,
                              void* d_out, int out_size, void* d_ws, size_t ws_size,
                              hipStream_t stream) {
  (void)in_sizes; (void)n_in; (void)out_size; (void)ws_size;
  const int*   mask = (const int*)d_in[0];    // [B,H,W] int32
  const float* vp   = (const float*)d_in[1];  // [B,2*VN,H,W] f32
  const int*   sidx = (const int*)d_in[2];    // [B,VN,NH,2] int32
  float* out = (float*)d_out;                 // [B,VN,2] f32

  // Workspace carve-out (~28 KB total)
  float* hyp_x = (float*)d_ws;
  float* hyp_y = hyp_x + B_ * VN_ * NH_;
  int*   hval  = (int*)(hyp_y + B_ * VN_ * NH_);

  int nhyp = B_ * VN_ * NH_;
  hyp_gen_kernel<<<(nhyp + 255) / 256, 256, 0, stream>>>(sidx, vp, hyp_x, hyp_y, hval);
  vote_refine_kernel<<<B_ * VN_, 256, 0, stream>>>(mask, vp, hyp_x, hyp_y, hval, out);
}